// BBoxEstimatorLoss_30159260353203
// MI455X (gfx1250) — compile-verified
//
#include <hip/hip_runtime.h>
#include <hip/hip_bf16.h>
#include <math.h>

typedef __attribute__((ext_vector_type(2))) float v2f;
typedef __attribute__((ext_vector_type(8))) float v8f;

#define BSZ      524288
#define NTHREADS 256
#define NBLK     (BSZ / NTHREADS)   // 2048, exact

__device__ __constant__ float c_mean_size[8][3] = {
    {3.88f, 1.63f, 1.53f}, {0.84f, 0.66f, 1.76f}, {1.76f, 0.60f, 1.74f},
    {16.17f, 2.58f, 3.23f}, {3.90f, 1.60f, 1.56f}, {1.73f, 0.58f, 1.37f},
    {0.91f, 0.48f, 1.78f}, {2.06f, 1.86f, 1.66f}};

__device__ __forceinline__ float huber_e(float a, float d) {
    // a >= 0 expected (caller passes |err| or a norm)
    float q = fminf(a, d);
    return 0.5f * q * q + d * (a - q);
}

// Sum `v` over all 32 lanes of the wave using V_WMMA_F32_16X16X4_F32.
// B = ones, C = 0  =>  D[m,n] = sum_k A[m,k].  Summing a lane's 8 D VGPRs
// gives sum over M=0..7 (lanes 0-15) / M=8..15 (lanes 16-31) for its column;
// adding lane L and L^16 yields the total sum of all A entries, independent
// of A's (lane,VGPR) -> (m,k) mapping. Requires EXEC == all ones: call in
// uniform control flow only.
__device__ __forceinline__ float wave_reduce_wmma(float v) {
    v2f a; a[0] = v;    a[1] = 0.0f;
    v2f b; b[0] = 1.0f; b[1] = 1.0f;
    v8f c = {};
    v8f d = __builtin_amdgcn_wmma_f32_16x16x4_f32(
        /*neg_a=*/false, a, /*neg_b=*/false, b,
        /*c_mod=*/(short)0, c, /*reuse_a=*/false, /*reuse_b=*/false);
    float t = ((d[0] + d[1]) + (d[2] + d[3])) + ((d[4] + d[5]) + (d[6] + d[7]));
    t += __shfl_xor(t, 16, 32);
    return t;  // every lane now holds the full wave sum
}

__global__ __launch_bounds__(NTHREADS) void bbox_loss_partial(
    const float* __restrict__ center,
    const float* __restrict__ center_label,
    const float* __restrict__ stage1_center,
    const float* __restrict__ heading_scores,
    const float* __restrict__ heading_residuals_normalized,
    const float* __restrict__ heading_residuals,
    const int*   __restrict__ heading_class_label,
    const float* __restrict__ heading_residuals_label,
    const float* __restrict__ size_scores,
    const float* __restrict__ size_residuals_normalized,
    const float* __restrict__ size_residuals,
    const int*   __restrict__ size_class_label,
    const float* __restrict__ size_residuals_label,
    float* __restrict__ partials) {
    const int i = blockIdx.x * NTHREADS + threadIdx.x;  // grid == BSZ exactly

    // ---- centers -------------------------------------------------------
    const float cx = center[3 * i + 0], cy = center[3 * i + 1], cz = center[3 * i + 2];
    const float lx = center_label[3 * i + 0], ly = center_label[3 * i + 1], lz = center_label[3 * i + 2];
    const float s1x = stage1_center[3 * i + 0], s1y = stage1_center[3 * i + 1], s1z = stage1_center[3 * i + 2];

    float dx = cx - lx, dy = cy - ly, dz = cz - lz;
    const float center_loss = huber_e(sqrtf(dx * dx + dy * dy + dz * dz), 2.0f);
    dx = cx - s1x; dy = cy - s1y; dz = cz - s1z;
    const float stage1_loss = huber_e(sqrtf(dx * dx + dy * dy + dz * dz), 1.0f);

    const int hcls = heading_class_label[i];
    const int scls = size_class_label[i];

    // ---- heading class loss: logsumexp(12) - score[hcls] ---------------
    const float4* hs4 = reinterpret_cast<const float4*>(heading_scores + 12 * (size_t)i);
    const float4 h0 = hs4[0], h1 = hs4[1], h2 = hs4[2];  // global_load_b128 x3
    float hm = fmaxf(fmaxf(fmaxf(h0.x, h0.y), fmaxf(h0.z, h0.w)),
                     fmaxf(fmaxf(fmaxf(h1.x, h1.y), fmaxf(h1.z, h1.w)),
                           fmaxf(fmaxf(h2.x, h2.y), fmaxf(h2.z, h2.w))));
    float hse = expf(h0.x - hm) + expf(h0.y - hm) + expf(h0.z - hm) + expf(h0.w - hm) +
                expf(h1.x - hm) + expf(h1.y - hm) + expf(h1.z - hm) + expf(h1.w - hm) +
                expf(h2.x - hm) + expf(h2.y - hm) + expf(h2.z - hm) + expf(h2.w - hm);
    const float hsel = heading_scores[12 * (size_t)i + hcls];
    const float heading_class_loss = (hm + logf(hse)) - hsel;

    // ---- size class loss: logsumexp(8) - score[scls] -------------------
    const float4* ss4 = reinterpret_cast<const float4*>(size_scores + 8 * (size_t)i);
    const float4 s0 = ss4[0], s1 = ss4[1];
    float sm = fmaxf(fmaxf(fmaxf(s0.x, s0.y), fmaxf(s0.z, s0.w)),
                     fmaxf(fmaxf(s1.x, s1.y), fmaxf(s1.z, s1.w)));
    float sse = expf(s0.x - sm) + expf(s0.y - sm) + expf(s0.z - sm) + expf(s0.w - sm) +
                expf(s1.x - sm) + expf(s1.y - sm) + expf(s1.z - sm) + expf(s1.w - sm);
    const float ssel = size_scores[8 * (size_t)i + scls];
    const float size_class_loss = (sm + logf(sse)) - ssel;

    // ---- heading residual loss -----------------------------------------
    const float hres_pred = heading_residuals_normalized[12 * (size_t)i + hcls];
    const float hres_lab  = heading_residuals_label[i] / (float)(M_PI / 12.0);
    const float heading_res_loss = huber_e(fabsf(hres_pred - hres_lab), 1.0f);

    // ---- size residual loss --------------------------------------------
    const float ms0 = c_mean_size[scls][0];
    const float ms1 = c_mean_size[scls][1];
    const float ms2 = c_mean_size[scls][2];
    const size_t sb = 24 * (size_t)i + 3 * scls;
    const float sp0 = size_residuals_normalized[sb + 0];
    const float sp1 = size_residuals_normalized[sb + 1];
    const float sp2 = size_residuals_normalized[sb + 2];
    const float srl0 = size_residuals_label[3 * i + 0];
    const float srl1 = size_residuals_label[3 * i + 1];
    const float srl2 = size_residuals_label[3 * i + 2];
    dx = srl0 / ms0 - sp0; dy = srl1 / ms1 - sp1; dz = srl2 / ms2 - sp2;
    const float size_res_loss = huber_e(sqrtf(dx * dx + dy * dy + dz * dz), 1.0f);

    // ---- corner loss ----------------------------------------------------
    const float bc = (float)hcls * (float)(2.0 * M_PI / 12.0);
    const float hp = heading_residuals[12 * (size_t)i + hcls] + bc;
    const float hl = heading_residuals_label[i] + bc;
    const float szp_l = ms0 + size_residuals[sb + 0];
    const float szp_w = ms1 + size_residuals[sb + 1];
    const float szp_h = ms2 + size_residuals[sb + 2];
    const float szl_l = ms0 + srl0;
    const float szl_w = ms1 + srl1;
    const float szl_h = ms2 + srl2;

    float cp, sp; __sincosf(0.0f, &sp, &cp);  // dummy init pattern avoided below
    cp = cosf(hp); sp = sinf(hp);
    const float cg = cosf(hl), sg = sinf(hl);

    const float SX[8] = {1.f, 1.f, -1.f, -1.f, 1.f, 1.f, -1.f, -1.f};
    const float SY[8] = {1.f, 1.f, 1.f, 1.f, -1.f, -1.f, -1.f, -1.f};
    const float SZ[8] = {1.f, -1.f, -1.f, 1.f, 1.f, -1.f, -1.f, 1.f};

    float corner_sum = 0.0f;
#pragma unroll
    for (int k = 0; k < 8; ++k) {
        // pred corner
        const float px = 0.5f * szp_l * SX[k];
        const float py = 0.5f * szp_h * SY[k];
        const float pz = 0.5f * szp_w * SZ[k];
        const float pxr = cp * px + sp * pz;
        const float pzr = -sp * px + cp * pz;
        const float PX = pxr + cx, PY = py + cy, PZ = pzr + cz;
        // gt corner (flip corner = negated rotation part: cos/sin(h+pi) = -cos/-sin)
        const float gx = 0.5f * szl_l * SX[k];
        const float gy = 0.5f * szl_h * SY[k];
        const float gz = 0.5f * szl_w * SZ[k];
        const float gxr = cg * gx + sg * gz;
        const float gzr = -sg * gx + cg * gz;
        const float GX = gxr + lx, GY = gy + ly, GZ = gzr + lz;
        const float FX = -gxr + lx, FZ = -gzr + lz;  // FY == GY
        const float e1x = PX - GX, e1y = PY - GY, e1z = PZ - GZ;
        const float e2x = PX - FX, e2z = PZ - FZ;
        const float d1 = e1x * e1x + e1y * e1y + e1z * e1z;
        const float d2 = e2x * e2x + e1y * e1y + e2z * e2z;
        corner_sum += huber_e(sqrtf(fminf(d1, d2)), 1.0f);
    }

    const float contrib = center_loss + stage1_loss + heading_class_loss + size_class_loss +
                          20.0f * heading_res_loss + 20.0f * size_res_loss +
                          10.0f * (corner_sum * 0.125f);

    // ---- block reduction: WMMA per wave (uniform EXEC), LDS across waves
    const float wsum = wave_reduce_wmma(contrib);
    __shared__ float sred[NTHREADS / 32];
    const int wave = threadIdx.x >> 5;
    const int lane = threadIdx.x & 31;
    if (lane == 0) sred[wave] = wsum;
    __syncthreads();
    if (threadIdx.x == 0) {
        float b = 0.0f;
#pragma unroll
        for (int w = 0; w < NTHREADS / 32; ++w) b += sred[w];
        partials[blockIdx.x] = b;
    }
}

__global__ __launch_bounds__(NTHREADS) void bbox_loss_final(
    const float* __restrict__ partials, float* __restrict__ out) {
    float s = 0.0f;
#pragma unroll
    for (int j = 0; j < NBLK / NTHREADS; ++j)
        s += partials[threadIdx.x + NTHREADS * j];
    const float wsum = wave_reduce_wmma(s);  // uniform: all 256 threads active
    __shared__ float sred[NTHREADS / 32];
    const int wave = threadIdx.x >> 5;
    const int lane = threadIdx.x & 31;
    if (lane == 0) sred[wave] = wsum;
    __syncthreads();
    if (threadIdx.x == 0) {
        float t = 0.0f;
#pragma unroll
        for (int w = 0; w < NTHREADS / 32; ++w) t += sred[w];
        out[0] = t * (1.0f / (float)BSZ);
    }
}

extern "C" void kernel_launch(void* const* d_in, const int* in_sizes, int n_in,
                              void* d_out, int out_size, void* d_ws, size_t ws_size,
                              hipStream_t stream) {
    (void)in_sizes; (void)n_in; (void)out_size; (void)ws_size;
    const float* center              = (const float*)d_in[0];
    const float* center_label        = (const float*)d_in[1];
    const float* stage1_center       = (const float*)d_in[2];
    const float* heading_scores      = (const float*)d_in[3];
    const float* heading_res_norm    = (const float*)d_in[4];
    const float* heading_residuals   = (const float*)d_in[5];
    const int*   heading_class_label = (const int*)d_in[6];
    const float* heading_res_label   = (const float*)d_in[7];
    const float* size_scores         = (const float*)d_in[8];
    const float* size_res_norm       = (const float*)d_in[9];
    const float* size_residuals      = (const float*)d_in[10];
    const int*   size_class_label    = (const int*)d_in[11];
    const float* size_res_label      = (const float*)d_in[12];

    float* partials = (float*)d_ws;  // NBLK floats = 8 KB

    bbox_loss_partial<<<NBLK, NTHREADS, 0, stream>>>(
        center, center_label, stage1_center, heading_scores, heading_res_norm,
        heading_residuals, heading_class_label, heading_res_label, size_scores,
        size_res_norm, size_residuals, size_class_label, size_res_label, partials);
    bbox_loss_final<<<1, NTHREADS, 0, stream>>>(partials, (float*)d_out);
}